// GCN_44676249813403
// MI455X (gfx1250) — compile-verified
//
#include <hip/hip_runtime.h>
#include <hip/hip_bf16.h>
#include <stdint.h>

typedef __attribute__((ext_vector_type(2))) float v2f;
typedef __attribute__((ext_vector_type(8))) float v8f;

#define N_NODES 100000
#define N_EDGES 3200000
#define F_IN    512
#define F_HID   16
#define F_OUT   32

// Fire-and-forget FP32 scatter-add: guarantee the native CDNA5 instruction
// (GLOBAL_ATOMIC_ADD_F32, no-return -> STOREcnt) instead of a possible CAS loop.
__device__ __forceinline__ void atomic_add_f32(float* p, float v) {
  asm volatile("global_atomic_add_f32 %0, %1, off" : : "v"(p), "v"(v) : "memory");
}

// ---------- degree / normalization ----------
__global__ void k_deg_init(float* __restrict__ deg) {
  int i = blockIdx.x * blockDim.x + threadIdx.x;
  if (i < N_NODES) deg[i] = 1.0f;   // self-loop contributes 1
}

__global__ void k_deg_count(const long long* __restrict__ dst, float* __restrict__ deg) {
  long long e = (long long)blockIdx.x * blockDim.x + threadIdx.x;
  if (e < N_EDGES) atomic_add_f32(&deg[dst[e]], 1.0f);
}

__global__ void k_dinv(float* __restrict__ deg) {
  int i = blockIdx.x * blockDim.x + threadIdx.x;
  if (i < N_NODES) deg[i] = rsqrtf(deg[i]);   // deg >= 1 always
}

// ---------- GEMM1: H1[100000x16] = X[100000x512] @ W1[512x16], f32 WMMA ----------
__global__ void __launch_bounds__(256) k_gemm1(const float* __restrict__ X,
                                               const float* __restrict__ W,
                                               float* __restrict__ H) {
  int wave = (int)((blockIdx.x * blockDim.x + threadIdx.x) >> 5);
  int lane = threadIdx.x & 31;
  if (wave >= N_NODES / 16) return;           // wave-uniform: EXEC all-1s inside
  int m  = lane & 15;                          // row (A) / col (B,C)
  int kh = lane >> 4;                          // K half-select
  const float* xp = X + (size_t)(wave * 16 + m) * F_IN + 2 * kh;
  const float* wp = W + 2 * kh * F_HID + m;
  v8f acc = {};
#pragma unroll 4
  for (int k = 0; k < F_IN; k += 4) {
    v2f a, b;
    a.x = xp[k];              a.y = xp[k + 1];             // A: M=m, K=k+2kh+{0,1}
    b.x = wp[k * F_HID];      b.y = wp[k * F_HID + F_HID]; // B: N=m, K=k+2kh+{0,1}
    acc = __builtin_amdgcn_wmma_f32_16x16x4_f32(false, a, false, b,
                                                (short)0, acc, false, false);
  }
  float* hp = H + (size_t)(wave * 16 + 8 * kh) * F_HID + m;
#pragma unroll
  for (int g = 0; g < 8; ++g) hp[g * F_HID] = acc[g];     // C/D: rows g, g+8
}

// ---------- self-loop init: agg1 = h1 * dinv^2 ----------
__global__ void k_selfloop16(const float* __restrict__ h, const float* __restrict__ dinv,
                             float* __restrict__ agg) {
  long long t = (long long)blockIdx.x * blockDim.x + threadIdx.x;
  if (t < (long long)N_NODES * F_HID) {
    float w = dinv[t >> 4];
    agg[t] = h[t] * w * w;
  }
}

// ---------- edge scatter layer1: 16 lanes per edge ----------
__global__ void k_scatter16(const long long* __restrict__ src, const long long* __restrict__ dst,
                            const float* __restrict__ h, const float* __restrict__ dinv,
                            float* __restrict__ agg) {
  long long t = (long long)blockIdx.x * blockDim.x + threadIdx.x;
  long long e = t >> 4;
  int n = (int)(t & 15);
  if (e >= N_EDGES) return;
  long long s = src[e], d = dst[e];
  float w = dinv[s] * dinv[d];
  atomic_add_f32(&agg[d * F_HID + n], h[s * F_HID + n] * w);
}

// ---------- GEMM2: H2[100000x32] = relu(agg1 + b1) @ W2[16x32], f32 WMMA ----------
__global__ void __launch_bounds__(256) k_gemm2(const float* __restrict__ A1,
                                               const float* __restrict__ B1,
                                               const float* __restrict__ W2,
                                               float* __restrict__ H2) {
  int wave = (int)((blockIdx.x * blockDim.x + threadIdx.x) >> 5);
  int lane = threadIdx.x & 31;
  if (wave >= N_NODES / 16) return;
  int m  = lane & 15;
  int kh = lane >> 4;
  const float* ap = A1 + (size_t)(wave * 16 + m) * F_HID + 2 * kh;
  v8f acc0 = {}, acc1 = {};
#pragma unroll
  for (int k = 0; k < F_HID; k += 4) {
    int k0 = k + 2 * kh;
    v2f a, b0, b1;
    a.x  = fmaxf(ap[k]     + B1[k0],     0.0f);   // fused bias + ReLU
    a.y  = fmaxf(ap[k + 1] + B1[k0 + 1], 0.0f);
    b0.x = W2[k0 * F_OUT + m];        b0.y = W2[(k0 + 1) * F_OUT + m];
    b1.x = W2[k0 * F_OUT + 16 + m];   b1.y = W2[(k0 + 1) * F_OUT + 16 + m];
    acc0 = __builtin_amdgcn_wmma_f32_16x16x4_f32(false, a, false, b0,
                                                 (short)0, acc0, false, false);
    acc1 = __builtin_amdgcn_wmma_f32_16x16x4_f32(false, a, false, b1,
                                                 (short)0, acc1, false, false);
  }
  float* hp = H2 + (size_t)(wave * 16 + 8 * kh) * F_OUT + m;
#pragma unroll
  for (int g = 0; g < 8; ++g) {
    hp[g * F_OUT]      = acc0[g];
    hp[g * F_OUT + 16] = acc1[g];
  }
}

// ---------- self-loop init: out = h2 * dinv^2 (also clears poison) ----------
__global__ void k_selfloop32(const float* __restrict__ h, const float* __restrict__ dinv,
                             float* __restrict__ out) {
  long long t = (long long)blockIdx.x * blockDim.x + threadIdx.x;
  if (t < (long long)N_NODES * F_OUT) {
    float w = dinv[t >> 5];
    out[t] = h[t] * w * w;
  }
}

// ---------- edge scatter layer2: 32 lanes (full wave) per edge ----------
__global__ void k_scatter32(const long long* __restrict__ src, const long long* __restrict__ dst,
                            const float* __restrict__ h, const float* __restrict__ dinv,
                            float* __restrict__ out) {
  long long t = (long long)blockIdx.x * blockDim.x + threadIdx.x;
  long long e = t >> 5;
  int n = (int)(t & 31);
  if (e >= N_EDGES) return;
  long long s = src[e], d = dst[e];
  float w = dinv[s] * dinv[d];
  atomic_add_f32(&out[d * F_OUT + n], h[s * F_OUT + n] * w);
}

// ---------- log-softmax over 32 classes, one wave per node, in place ----------
__global__ void k_logsoftmax(float* __restrict__ out, const float* __restrict__ b2) {
  long long t = (long long)blockIdx.x * blockDim.x + threadIdx.x;
  long long i = t >> 5;
  int n = (int)(t & 31);
  if (i >= N_NODES) return;
  float z = out[i * F_OUT + n] + b2[n];
  float mx = z;
#pragma unroll
  for (int off = 16; off > 0; off >>= 1) mx = fmaxf(mx, __shfl_xor(mx, off, 32));
  float ex = expf(z - mx);
  float s = ex;
#pragma unroll
  for (int off = 16; off > 0; off >>= 1) s += __shfl_xor(s, off, 32);
  out[i * F_OUT + n] = z - mx - logf(s);
}

extern "C" void kernel_launch(void* const* d_in, const int* in_sizes, int n_in,
                              void* d_out, int out_size, void* d_ws, size_t ws_size,
                              hipStream_t stream) {
  const float*     X   = (const float*)d_in[0];
  const long long* ei  = (const long long*)d_in[1];
  const long long* src = ei;
  const long long* dst = ei + N_EDGES;
  const float*     W1  = (const float*)d_in[2];
  const float*     b1  = (const float*)d_in[3];
  const float*     W2  = (const float*)d_in[4];
  const float*     b2  = (const float*)d_in[5];
  float* out = (float*)d_out;

  char* ws = (char*)d_ws;
  float* dinv = (float*)ws;                                    // 400 KB (deg -> dinv in place)
  float* h1   = (float*)(ws + (1 << 19));                      // 6.4 MB
  float* agg1 = (float*)(ws + (1 << 19) + 6400000);            // 6.4 MB
  float* h2   = (float*)(ws + (1 << 19) + 12800000);           // 12.8 MB

  const int B = 256;
  k_deg_init  <<<(N_NODES + B - 1) / B, B, 0, stream>>>(dinv);
  k_deg_count <<<(N_EDGES + B - 1) / B, B, 0, stream>>>(dst, dinv);
  k_dinv      <<<(N_NODES + B - 1) / B, B, 0, stream>>>(dinv);

  int tiles = N_NODES / 16;                                    // 6250 waves
  k_gemm1 <<<(tiles * 32 + B - 1) / B, B, 0, stream>>>(X, W1, h1);

  k_selfloop16 <<<(int)(((long long)N_NODES * F_HID + B - 1) / B), B, 0, stream>>>(h1, dinv, agg1);
  k_scatter16  <<<(int)(((long long)N_EDGES * 16 + B - 1) / B), B, 0, stream>>>(src, dst, h1, dinv, agg1);

  k_gemm2 <<<(tiles * 32 + B - 1) / B, B, 0, stream>>>(agg1, b1, W2, h2);

  k_selfloop32 <<<(int)(((long long)N_NODES * F_OUT + B - 1) / B), B, 0, stream>>>(h2, dinv, out);
  k_scatter32  <<<(int)(((long long)N_EDGES * 32 + B - 1) / B), B, 0, stream>>>(src, dst, h2, dinv, out);

  k_logsoftmax <<<(int)(((long long)N_NODES * F_OUT + B - 1) / B), B, 0, stream>>>(out, b2);
}